// GCN_32607391711999
// MI455X (gfx1250) — compile-verified
//
#include <hip/hip_runtime.h>

// ---------------------------------------------------------------------------
// GCN / ChebConv(K=3) on MI455X (gfx1250).
// Layer 1 collapses to SpMM + elementwise because x == I.
// Layer 2 is a fused  [h | T1 | T2](Nx384) @ [W2](384x128)  GEMM done with
// V_WMMA_F32_16X16X4_F32 (fp32-exact, matches the fp32 reference).
// GEMM v2: A strips staged in LDS (shared by all 8 waves of the block),
// inner loop = ds_load_b64 + 2x global_load_b32 with immediate offsets.
// ---------------------------------------------------------------------------

typedef __attribute__((ext_vector_type(2))) float v2f;
typedef __attribute__((ext_vector_type(8))) float v8f;

// ---------------- utility kernels ----------------

__global__ void zero_f32(float* __restrict__ p, int n4) {
    int i = blockIdx.x * blockDim.x + threadIdx.x;
    if (i < n4) ((float4*)p)[i] = make_float4(0.f, 0.f, 0.f, 0.f);
}

__global__ void deg_kernel(const int* __restrict__ row,
                           const float* __restrict__ ew,
                           float* __restrict__ deg, int E) {
    int e = blockIdx.x * blockDim.x + threadIdx.x;
    if (e < E) atomicAdd(&deg[row[e]], ew[e]);
}

__global__ void dinv_kernel(const float* __restrict__ deg,
                            float* __restrict__ dinv, int N) {
    int i = blockIdx.x * blockDim.x + threadIdx.x;
    if (i < N) {
        float d = deg[i];
        dinv[i] = (d > 0.f) ? rsqrtf(d) : 0.f;
    }
}

__global__ void normw_kernel(const int* __restrict__ row,
                             const int* __restrict__ col,
                             const float* __restrict__ ew,
                             const float* __restrict__ dinv,
                             float* __restrict__ nw, int E) {
    int e = blockIdx.x * blockDim.x + threadIdx.x;
    if (e < E) nw[e] = -dinv[row[e]] * ew[e] * dinv[col[e]];
}

// out[col_e, :] += nw_e * in[row_e, :]   (scatter SpMM, D=128)
// one thread per (edge, group-of-4-features)
__global__ void spmm_kernel(const int* __restrict__ row,
                            const int* __restrict__ col,
                            const float* __restrict__ nw,
                            const float* __restrict__ in,
                            float* __restrict__ out,
                            int E, int D) {
    int t = blockIdx.x * blockDim.x + threadIdx.x;
    int groups = D >> 2;             // 32
    int e = t / groups;
    int g = t - e * groups;
    if (e >= E) return;
    int r = row[e], c = col[e];
    float w = nw[e];
    float4 v = ((const float4*)(in + (size_t)r * D))[g];
    float* dst = out + (size_t)c * D + (g << 2);
    atomicAdd(dst + 0, w * v.x);
    atomicAdd(dst + 1, w * v.y);
    atomicAdd(dst + 2, w * v.z);
    atomicAdd(dst + 3, w * v.w);
}

// h = relu(W1_0 + S1 + 2*S2 - W1_2 + b1)   (elementwise over N*D, float4)
__global__ void layer1_combine(const float* __restrict__ w10,
                               const float* __restrict__ s1,
                               const float* __restrict__ s2,
                               const float* __restrict__ w12,
                               const float* __restrict__ b1,
                               float* __restrict__ h,
                               int n4, int D) {
    int i = blockIdx.x * blockDim.x + threadIdx.x;
    if (i >= n4) return;
    int cg = i & ((D >> 2) - 1);     // float4 group within a row
    float4 a = ((const float4*)w10)[i];
    float4 s = ((const float4*)s1)[i];
    float4 t = ((const float4*)s2)[i];
    float4 m = ((const float4*)w12)[i];
    float4 b = ((const float4*)b1)[cg];
    float4 o;
    o.x = fmaxf(a.x + s.x + 2.f * t.x - m.x + b.x, 0.f);
    o.y = fmaxf(a.y + s.y + 2.f * t.y - m.y + b.y, 0.f);
    o.z = fmaxf(a.z + s.z + 2.f * t.z - m.z + b.z, 0.f);
    o.w = fmaxf(a.w + s.w + 2.f * t.w - m.w + b.w, 0.f);
    ((float4*)h)[i] = o;
}

// T2 = 2*U - h
__global__ void t2_combine(const float* __restrict__ u,
                           const float* __restrict__ h,
                           float* __restrict__ t2, int n4) {
    int i = blockIdx.x * blockDim.x + threadIdx.x;
    if (i >= n4) return;
    float4 a = ((const float4*)u)[i];
    float4 b = ((const float4*)h)[i];
    float4 o;
    o.x = 2.f * a.x - b.x;  o.y = 2.f * a.y - b.y;
    o.z = 2.f * a.z - b.z;  o.w = 2.f * a.w - b.w;
    ((float4*)t2)[i] = o;
}

// ---------------- fused layer-2 WMMA GEMM (LDS-staged A) ----------------
// out(N x 128) = A0@B0 + A1@B1 + A2@B2 + bias,  A* are N x 128, B* 128 x 128.
// One block = one 16-row output strip (tm = blockIdx.x); wave w = column
// tile tn = w. All 8 waves share the same A rows -> stage the three 16x128
// A strips in LDS (rows padded to 132 floats => conflict-free 16-lane column
// reads). Inner loop: ds_load_b64 (A pair) + 2x global_load_b32 (B, L2-hot)
// with immediate offsets; V_WMMA_F32_16X16X4_F32 accumulate, K-step 4.
// f32 layouts (ISA 7.12.2): lane = l&15 indexes M (A) / N (B,C,D); l>>4
// picks the K half:  A vgpr j -> K = k0 + 2*half + j ; B vgpr j -> same K,
// N = lane ;  C/D vgpr r -> M = r + 8*half, N = lane.
#define DC 128
#define APAD 132

__global__ void __launch_bounds__(256)
gemm3_wmma_f32(const float* __restrict__ A0,
               const float* __restrict__ A1,
               const float* __restrict__ A2,
               const float* __restrict__ B,   // W2: 3 x 128 x 128 contiguous
               const float* __restrict__ bias,
               float* __restrict__ out) {
    __shared__ float As[3][16][APAD];          // 25,344 B

    const int wave = threadIdx.x >> 5;
    const int lane = threadIdx.x & 31;
    const int idx  = lane & 15;
    const int half = lane >> 4;

    const int tm   = blockIdx.x;               // row tile
    const int tn   = wave;                     // column tile (0..7)
    const int colB = tn * 16 + idx;

    const float* Am[3] = {A0, A1, A2};

    // cooperative stage of the three 16x128 A strips (float4 loads)
#pragma unroll
    for (int m = 0; m < 3; ++m) {
        const float* src = Am[m] + (size_t)tm * 16 * DC;
#pragma unroll
        for (int i = 0; i < 2; ++i) {
            int f  = threadIdx.x + i * 256;    // 0..511 float4 slots
            int r  = f >> 5;                   // 32 float4 per row
            int c4 = f & 31;
            float4 v = ((const float4*)(src + (size_t)r * DC))[c4];
            *(float4*)&As[m][r][c4 << 2] = v;
        }
    }
    __syncthreads();

    v8f c = {0.f, 0.f, 0.f, 0.f, 0.f, 0.f, 0.f, 0.f};

#pragma unroll
    for (int m = 0; m < 3; ++m) {
        const float* Arow  = &As[m][idx][2 * half];             // LDS base
        const float* Bbase = B + (size_t)m * DC * DC            // B[2*half][colB]
                               + (size_t)2 * half * DC + colB;
#pragma unroll
        for (int k0 = 0; k0 < DC; k0 += 4) {
            v2f a = *(const v2f*)(Arow + k0);                   // ds_load_b64
            v2f b;
            b[0] = Bbase[(size_t)k0 * DC];                      // imm offset
            b[1] = Bbase[(size_t)k0 * DC + DC];                 // imm offset
            c = __builtin_amdgcn_wmma_f32_16x16x4_f32(
                    false, a, false, b, (short)0, c, false, false);
        }
    }

    float bv = bias[colB];
#pragma unroll
    for (int r = 0; r < 8; ++r) {
        int orow = tm * 16 + r + 8 * half;
        out[(size_t)orow * DC + colB] = c[r] + bv;
    }
}

// ---------------- host-side launcher ----------------

extern "C" void kernel_launch(void* const* d_in, const int* in_sizes, int n_in,
                              void* d_out, int out_size, void* d_ws, size_t ws_size,
                              hipStream_t stream) {
    // inputs: x(unused), edge_index, edge_weight, W1, b1, W2, b2
    const int*   ei  = (const int*)d_in[1];
    const float* ew  = (const float*)d_in[2];
    const float* W1  = (const float*)d_in[3];
    const float* b1  = (const float*)d_in[4];
    const float* W2  = (const float*)d_in[5];
    const float* b2  = (const float*)d_in[6];
    float* out = (float*)d_out;

    const int E = in_sizes[2];
    const int D = in_sizes[4];                 // 128
    const int N = in_sizes[3] / (3 * D);       // 4096
    const int ND = N * D;

    const int* row = ei;
    const int* col = ei + E;
    const float* W1_0 = W1;
    const float* W1_1 = W1 + (size_t)ND;
    const float* W1_2 = W1 + (size_t)2 * ND;

    // workspace carve-up (all 16B aligned)
    char* ws = (char*)d_ws;
    float* deg   = (float*)ws;                   ws += (size_t)N * 4;
    float* dinv  = (float*)ws;                   ws += (size_t)N * 4;
    float* normw = (float*)ws;                   ws += (size_t)E * 4;
    float* B0 = (float*)ws;                      ws += (size_t)ND * 4;  // S1 / T1
    float* B1 = (float*)ws;                      ws += (size_t)ND * 4;  // S2a / U
    float* B2 = (float*)ws;                      ws += (size_t)ND * 4;  // S2 / T2
    float* H  = (float*)ws;                      ws += (size_t)ND * 4;  // h

    const int TB = 256;
    const int nd4 = ND / 4;
    dim3 gZero((nd4 + TB - 1) / TB), gN4(((N / 4) + TB - 1) / TB);
    dim3 gE((E + TB - 1) / TB), gNn((N + TB - 1) / TB);
    const int spmmThreads = E * (D / 4);
    dim3 gSpmm((spmmThreads + TB - 1) / TB);

    // --- normalization ---
    zero_f32<<<gN4, TB, 0, stream>>>(deg, N / 4);
    deg_kernel<<<gE, TB, 0, stream>>>(row, ew, deg, E);
    dinv_kernel<<<gNn, TB, 0, stream>>>(deg, dinv, N);
    normw_kernel<<<gE, TB, 0, stream>>>(row, col, ew, dinv, normw, E);

    // --- layer 1 (x == I, so no GEMM needed) ---
    zero_f32<<<gZero, TB, 0, stream>>>(B0, nd4);
    spmm_kernel<<<gSpmm, TB, 0, stream>>>(row, col, normw, W1_1, B0, E, D); // S1
    zero_f32<<<gZero, TB, 0, stream>>>(B1, nd4);
    spmm_kernel<<<gSpmm, TB, 0, stream>>>(row, col, normw, W1_2, B1, E, D); // S2a
    zero_f32<<<gZero, TB, 0, stream>>>(B2, nd4);
    spmm_kernel<<<gSpmm, TB, 0, stream>>>(row, col, normw, B1, B2, E, D);   // S2
    layer1_combine<<<gZero, TB, 0, stream>>>(W1_0, B0, B2, W1_2, b1, H, nd4, D);

    // --- layer 2 ---
    zero_f32<<<gZero, TB, 0, stream>>>(B0, nd4);
    spmm_kernel<<<gSpmm, TB, 0, stream>>>(row, col, normw, H, B0, E, D);    // T1
    zero_f32<<<gZero, TB, 0, stream>>>(B1, nd4);
    spmm_kernel<<<gSpmm, TB, 0, stream>>>(row, col, normw, B0, B1, E, D);   // U
    t2_combine<<<gZero, TB, 0, stream>>>(B1, H, B2, nd4);                   // T2

    // fused GEMM: out = H@W2[0] + T1@W2[1] + T2@W2[2] + b2
    gemm3_wmma_f32<<<N / 16, 256, 0, stream>>>(H, B0, B2, W2, b2, out);
}